// FraudDetectionModel_65481071408354
// MI455X (gfx1250) — compile-verified
//
#include <hip/hip_runtime.h>
#include <hip/hip_bf16.h>

typedef __attribute__((ext_vector_type(2))) float v2f;
typedef __attribute__((ext_vector_type(8))) float v8f;

__device__ __forceinline__ float fast_tanh(float v) {
#if __has_builtin(__builtin_amdgcn_tanhf)
    return __builtin_amdgcn_tanhf(v);
#elif __has_builtin(__builtin_amdgcn_tanh_f32)
    return __builtin_amdgcn_tanh_f32(v);
#else
    return tanhf(v);
#endif
}

// ---------------------------------------------------------------------------
// Prep kernel: fold scale/shift into next layer's weights and emit them in the
// exact WMMA A-matrix (16x4 f32) lane layout:
//   a.x : lane m (m<16) = A[m][0] ; lane 16+m = A[m][2] (bias column)
//   a.y : lane m        = A[m][1] ; lane 16+m = A[m][3] (= 0)
// Only rows 0,1 of A are non-zero.  ws[2048..2050] = folded output params.
// ---------------------------------------------------------------------------
__global__ void fraud_prep(const float* __restrict__ W, const float* __restrict__ b,
                           const float* __restrict__ scale, const float* __restrict__ shift,
                           const float* __restrict__ w_out, const float* __restrict__ b_out,
                           float* __restrict__ ws) {
    const int t = threadIdx.x;          // 0..1023
    const int l = t >> 5;               // layer
    const int lane = t & 31;

    const float s0 = (l > 0) ? scale[(l - 1) * 2 + 0] : 1.0f;
    const float s1 = (l > 0) ? scale[(l - 1) * 2 + 1] : 1.0f;
    const float h0 = (l > 0) ? shift[(l - 1) * 2 + 0] : 0.0f;
    const float h1 = (l > 0) ? shift[(l - 1) * 2 + 1] : 0.0f;

    const float W00 = W[l * 4 + 0], W01 = W[l * 4 + 1];
    const float W10 = W[l * 4 + 2], W11 = W[l * 4 + 3];

    const float w00 = W00 * s0, w01 = W01 * s1;
    const float w10 = W10 * s0, w11 = W11 * s1;
    const float bf0 = b[l * 2 + 0] + W00 * h0 + W01 * h1;
    const float bf1 = b[l * 2 + 1] + W10 * h0 + W11 * h1;

    float a0 = 0.0f, a1 = 0.0f;
    if (lane == 0)       { a0 = w00; a1 = w01; }
    else if (lane == 1)  { a0 = w10; a1 = w11; }
    else if (lane == 16) { a0 = bf0; }
    else if (lane == 17) { a0 = bf1; }

    ws[(l * 32 + lane) * 2 + 0] = a0;
    ws[(l * 32 + lane) * 2 + 1] = a1;

    if (t == 0) {
        ws[2048] = w_out[0] * scale[31 * 2 + 0];
        ws[2049] = w_out[1] * scale[31 * 2 + 1];
        ws[2050] = b_out[0] + w_out[0] * shift[31 * 2 + 0] + w_out[1] * shift[31 * 2 + 1];
    }
}

// ---------------------------------------------------------------------------
// Main kernel: 32 batch rows per wave, two interleaved 16-row WMMA chains.
// B-matrix (4x16):  b.x lanes 0-15 = act0 (K=0), lanes 16-31 = 1.0 (K=2)
//                   b.y lanes 0-15 = act1 (K=1), lanes 16-31 = 0.0 (K=3)
// D rows 0/1 return z0/z1 for item n in lane n -> tanh -> next B. No shuffles.
// ---------------------------------------------------------------------------
__global__ __launch_bounds__(256) void fraud_main(const float* __restrict__ x,
                                                  const float* __restrict__ af,
                                                  float* __restrict__ out) {
    const int lane = threadIdx.x & 31;
    const int base = blockIdx.x * 256 + (threadIdx.x >> 5) * 32;   // 32 items / wave
    const bool lo = lane < 16;

    // Preload all 32 layers of A-matrix fragments (wave-uniform across grid, L2-hot).
    v2f a[32];
#pragma unroll
    for (int l = 0; l < 32; ++l)
        a[l] = ((const v2f*)af)[l * 32 + lane];

    const float wo0 = af[2048];
    const float wo1 = af[2049];
    const float bo  = af[2050];

    // Coalesced float2 loads: chain A = items base..base+15, chain B = +16..+31.
    float uA0 = 0.0f, uA1 = 0.0f, uB0 = 0.0f, uB1 = 0.0f;
    if (lo) {
        const float2 va = ((const float2*)x)[base + lane];
        const float2 vb = ((const float2*)x)[base + 16 + lane];
        uA0 = va.x; uA1 = va.y;
        uB0 = vb.x; uB1 = vb.y;
    }

    const v8f cz = {0.0f, 0.0f, 0.0f, 0.0f, 0.0f, 0.0f, 0.0f, 0.0f};

#pragma unroll
    for (int l = 0; l < 32; ++l) {
        v2f bA, bB;
        bA.x = lo ? uA0 : 1.0f;   // K=2 row of B must be all-ones (bias column)
        bA.y = uA1;               // K=3 row already 0 (tanh(0)=0 / zero-init)
        bB.x = lo ? uB0 : 1.0f;
        bB.y = uB1;

        v8f dA = __builtin_amdgcn_wmma_f32_16x16x4_f32(false, a[l], false, bA,
                                                       (short)0, cz, false, false);
        v8f dB = __builtin_amdgcn_wmma_f32_16x16x4_f32(false, a[l], false, bB,
                                                       (short)0, cz, false, false);

        uA0 = fast_tanh(dA[0]);   // M=0 row: z0 per item (high lanes stay 0)
        uA1 = fast_tanh(dA[1]);   // M=1 row: z1 per item
        uB0 = fast_tanh(dB[0]);
        uB1 = fast_tanh(dB[1]);
    }

    const float rA = fmaf(wo0, uA0, fmaf(wo1, uA1, bo));
    const float rB = fmaf(wo0, uB0, fmaf(wo1, uB1, bo));
    if (lo) {
        out[base + lane]      = rA;
        out[base + 16 + lane] = rB;
    }
}

extern "C" void kernel_launch(void* const* d_in, const int* in_sizes, int n_in,
                              void* d_out, int out_size, void* d_ws, size_t ws_size,
                              hipStream_t stream) {
    const float* x     = (const float*)d_in[0];
    const float* W     = (const float*)d_in[1];
    const float* b     = (const float*)d_in[2];
    const float* scale = (const float*)d_in[3];
    const float* shift = (const float*)d_in[4];
    const float* w_out = (const float*)d_in[5];
    const float* b_out = (const float*)d_in[6];
    float* out = (float*)d_out;
    float* ws  = (float*)d_ws;

    fraud_prep<<<1, 1024, 0, stream>>>(W, b, scale, shift, w_out, b_out, ws);

    const int n = in_sizes[0] / 2;              // B rows (8,388,608)
    fraud_main<<<n / 256, 256, 0, stream>>>(x, ws, out);
}